// GraphEmbedding2_23570780521026
// MI455X (gfx1250) — compile-verified
//
#include <hip/hip_runtime.h>
#include <hip/hip_bf16.h>

#define DIM 128

typedef float v2f __attribute__((ext_vector_type(2)));
typedef float v8f __attribute__((ext_vector_type(8)));

__global__ void fill_zero_kernel(float* __restrict__ p, int n) {
  int i = blockIdx.x * blockDim.x + threadIdx.x;
  if (i < n) p[i] = 0.0f;
}

__global__ void degree_kernel(const int* __restrict__ src, const int* __restrict__ dst,
                              float* __restrict__ outdeg, float* __restrict__ indeg, int nE) {
  int e = blockIdx.x * blockDim.x + threadIdx.x;
  if (e < nE) {
    atomicAdd(&outdeg[src[e]], 1.0f);
    atomicAdd(&indeg[dst[e]], 1.0f);
  }
}

__global__ void norm_kernel(float* __restrict__ outdeg, float* __restrict__ indeg, int n) {
  int i = blockIdx.x * blockDim.x + threadIdx.x;
  if (i < n) {
    outdeg[i] = rsqrtf(fmaxf(outdeg[i], 1.0f));
    indeg[i]  = rsqrtf(fmaxf(indeg[i], 1.0f));
  }
}

// One wave32 per edge; each lane moves a float4 (32 lanes * 4 = 128 features).
// src/dst are wave-uniform -> scalarized loads. norm_src scaling fused here.
__global__ void scatter_kernel(const float* __restrict__ x, const int* __restrict__ src,
                               const int* __restrict__ dst, const float* __restrict__ norm_src,
                               float* __restrict__ agg, int nE) {
  int tid  = blockIdx.x * blockDim.x + threadIdx.x;
  int e    = tid >> 5;
  int lane = tid & 31;
  if (e >= nE) return;
  int s = src[e];
  int d = dst[e];
  float ns = norm_src[s];
  float4 v = ((const float4*)(x + (size_t)s * DIM))[lane];
  float* o = agg + (size_t)d * DIM + lane * 4;
  atomicAdd(o + 0, v.x * ns);
  atomicAdd(o + 1, v.y * ns);
  atomicAdd(o + 2, v.z * ns);
  atomicAdd(o + 3, v.w * ns);
}

// out[n,:] = relu((agg[n,:] * norm_dst[n]) @ W + b)
// 8 waves / block; each wave: 16-row strip x all 128 cols via 8 v8f accumulators.
// W staged transposed in LDS with XOR swizzle so B fetches are conflict-free ds_load_b64.
__global__ __launch_bounds__(256)
void gemm_relu_kernel(const float* __restrict__ agg, const float* __restrict__ norm_dst,
                      const float* __restrict__ W, const float* __restrict__ bias,
                      float* __restrict__ out, int nNodes) {
  __shared__ float Wt[DIM * DIM];  // Wt[c][k] = W[k][c], k XOR-swizzled by col
  int t = threadIdx.x;
  #pragma unroll
  for (int i = 0; i < (DIM * DIM) / 256; ++i) {
    int id = i * 256 + t;
    int k = id >> 7;
    int c = id & 127;
    Wt[c * DIM + (k ^ ((c & 7) << 3))] = W[id];
  }
  __syncthreads();

  int wave = t >> 5;
  int lane = t & 31;
  int tile = blockIdx.x * 8 + wave;          // 16-row tile index
  if (tile * 16 >= nNodes) return;           // wave-uniform

  int m     = lane & 15;                     // N index / row-within-tile for A
  int khalf = lane >> 4;                     // 0 or 1
  int row   = tile * 16 + m;
  bool rvalid = row < nNodes;
  float nd = rvalid ? norm_dst[row] : 0.0f;
  const float* arow = agg + (size_t)(rvalid ? row : 0) * DIM;

  v8f acc[8];
  #pragma unroll
  for (int nt = 0; nt < 8; ++nt) { v8f z = {}; acc[nt] = z; }

  for (int k0 = 0; k0 < DIM; k0 += 4) {
    int ka = k0 + 2 * khalf;                 // A VGPR0/1 hold K = {0,2}/{1,3}
    float2 av = *(const float2*)(arow + ka);
    v2f a;
    a.x = av.x * nd;
    a.y = av.y * nd;
    #pragma unroll
    for (int nt = 0; nt < 8; ++nt) {
      int col = nt * 16 + m;
      float2 bv = *(const float2*)&Wt[col * DIM + (ka ^ ((col & 7) << 3))];
      v2f b;
      b.x = bv.x;                            // B row K=ka   (per-lane-half)
      b.y = bv.y;                            // B row K=ka+1
      acc[nt] = __builtin_amdgcn_wmma_f32_16x16x4_f32(
          false, a, false, b, (short)0, acc[nt], false, false);
    }
  }

  // C/D layout: VGPR r -> M = r + 8*khalf, N = lane&15. Fuse bias + relu.
  #pragma unroll
  for (int nt = 0; nt < 8; ++nt) {
    int col = nt * 16 + m;
    float bv = bias[col];
    #pragma unroll
    for (int r = 0; r < 8; ++r) {
      int orow = tile * 16 + r + 8 * khalf;
      if (orow < nNodes)
        out[(size_t)orow * DIM + col] = fmaxf(acc[nt][r] + bv, 0.0f);
    }
  }
}

__global__ void pool_sum_kernel(const float* __restrict__ x, const int* __restrict__ gids,
                                float* __restrict__ gsum, float* __restrict__ gcnt, int n) {
  int tid  = blockIdx.x * blockDim.x + threadIdx.x;
  int node = tid >> 5;
  int lane = tid & 31;
  if (node >= n) return;
  int g = gids[node];
  float4 v = ((const float4*)(x + (size_t)node * DIM))[lane];
  float* o = gsum + g * DIM + lane * 4;
  atomicAdd(o + 0, v.x);
  atomicAdd(o + 1, v.y);
  atomicAdd(o + 2, v.z);
  atomicAdd(o + 3, v.w);
  if (lane == 0) atomicAdd(&gcnt[g], 1.0f);
}

__global__ void pool_div_kernel(const float* __restrict__ gsum, const float* __restrict__ gcnt,
                                float* __restrict__ out, int total) {
  int i = blockIdx.x * blockDim.x + threadIdx.x;
  if (i < total) out[i] = gsum[i] / fmaxf(gcnt[i >> 7], 1.0f);
}

extern "C" void kernel_launch(void* const* d_in, const int* in_sizes, int n_in,
                              void* d_out, int out_size, void* d_ws, size_t ws_size,
                              hipStream_t stream) {
  const float* h  = (const float*)d_in[0];
  const int* src  = (const int*)d_in[1];
  const int* dst  = (const int*)d_in[2];
  const int* gids = (const int*)d_in[3];
  const float* W1 = (const float*)d_in[4];
  const float* b1 = (const float*)d_in[5];
  const float* W2 = (const float*)d_in[6];
  const float* b2 = (const float*)d_in[7];
  const float* W3 = (const float*)d_in[8];
  const float* b3 = (const float*)d_in[9];

  const int nE = in_sizes[1];   // 600000
  const int nN = in_sizes[3];   // 50000 (graph_ids length)
  const int G  = 64;

  float* ws = (float*)d_ws;
  float* norm_src = ws;  ws += nN;              // also outdeg during accumulation
  float* norm_dst = ws;  ws += nN;              // also indeg
  float* bufA = ws;      ws += (size_t)nN * DIM;
  float* bufB = ws;      ws += (size_t)nN * DIM;
  float* agg  = ws;      ws += (size_t)nN * DIM;
  float* gsum = ws;      ws += G * DIM;
  float* gcnt = ws;      ws += G;

  const int B = 256;
  const int aggN = nN * DIM;
  const int nTiles = (nN + 15) / 16;
  const int gemmBlocks = (nTiles + 7) / 8;
  const int scatterBlocks = (nE * 32 + B - 1) / B;

  // degrees -> norms (norm_src/norm_dst are contiguous: one fill covers both)
  fill_zero_kernel<<<(2 * nN + B - 1) / B, B, 0, stream>>>(norm_src, 2 * nN);
  degree_kernel<<<(nE + B - 1) / B, B, 0, stream>>>(src, dst, norm_src, norm_dst, nE);
  norm_kernel<<<(nN + B - 1) / B, B, 0, stream>>>(norm_src, norm_dst, nN);

  // layer 1: h -> bufA
  fill_zero_kernel<<<(aggN + B - 1) / B, B, 0, stream>>>(agg, aggN);
  scatter_kernel<<<scatterBlocks, B, 0, stream>>>(h, src, dst, norm_src, agg, nE);
  gemm_relu_kernel<<<gemmBlocks, B, 0, stream>>>(agg, norm_dst, W1, b1, bufA, nN);

  // layer 2: bufA -> bufB
  fill_zero_kernel<<<(aggN + B - 1) / B, B, 0, stream>>>(agg, aggN);
  scatter_kernel<<<scatterBlocks, B, 0, stream>>>(bufA, src, dst, norm_src, agg, nE);
  gemm_relu_kernel<<<gemmBlocks, B, 0, stream>>>(agg, norm_dst, W2, b2, bufB, nN);

  // layer 3: bufB -> bufA
  fill_zero_kernel<<<(aggN + B - 1) / B, B, 0, stream>>>(agg, aggN);
  scatter_kernel<<<scatterBlocks, B, 0, stream>>>(bufB, src, dst, norm_src, agg, nE);
  gemm_relu_kernel<<<gemmBlocks, B, 0, stream>>>(agg, norm_dst, W3, b3, bufA, nN);

  // per-graph mean pooling (gsum/gcnt contiguous: one fill covers both)
  fill_zero_kernel<<<(G * DIM + G + B - 1) / B, B, 0, stream>>>(gsum, G * DIM + G);
  pool_sum_kernel<<<(nN * 32 + B - 1) / B, B, 0, stream>>>(bufA, gids, gsum, gcnt, nN);
  pool_div_kernel<<<(G * DIM + B - 1) / B, B, 0, stream>>>(gsum, gcnt, (float*)d_out, G * DIM);
}